// GIN_65171833750208
// MI455X (gfx1250) — compile-verified
//
#include <hip/hip_runtime.h>

// GIN forward for MI455X (gfx1250, wave32).
// fp32 end-to-end; GEMMs use V_WMMA_F32_16X16X4_F32 (true fp32 matrix op).
// W-tile staging into LDS uses the Tensor Data Mover (TENSOR_LOAD_TO_LDS)
// with descriptor padding reproducing the bank-decorrelated stride-72 layout.

typedef float v2f __attribute__((ext_vector_type(2)));
typedef float v8f __attribute__((ext_vector_type(8)));
typedef unsigned int u32x4 __attribute__((ext_vector_type(4)));
typedef int i32x4 __attribute__((ext_vector_type(4)));
typedef int i32x8 __attribute__((ext_vector_type(8)));

#define LDS_STRIDE 72   // 72 mod 32 == 8 -> the two K-half lane groups hit disjoint LDS bank sets

// LDS byte offset of a __shared__ object (addrspace(3) ptrtoint).
__device__ static inline unsigned gin_lds_off(void* p) {
    return (unsigned)(unsigned long long)(__attribute__((address_space(3))) void*)p;
}

// ---------------------------------------------------------------------------
// Fold linear bias + BatchNorm (eval) into per-column scale/shift.
// P layout per layer: [s1(128) | t1(128) | s2(128) | t2(128)]
// ---------------------------------------------------------------------------
__global__ void gin_fold_bn(const float* __restrict__ b1,
                            const float* __restrict__ g1, const float* __restrict__ bb1,
                            const float* __restrict__ m1, const float* __restrict__ v1,
                            const float* __restrict__ b2,
                            const float* __restrict__ g2, const float* __restrict__ bb2,
                            const float* __restrict__ m2, const float* __restrict__ v2,
                            float* __restrict__ P, int LH) {
    int t = blockIdx.x * blockDim.x + threadIdx.x;
    if (t >= LH) return;
    int layer = t >> 7;
    int c     = t & 127;
    float s1 = g1[t] * rsqrtf(v1[t] + 1e-5f);
    float t1 = (b1[t] - m1[t]) * s1 + bb1[t];
    float s2 = g2[t] * rsqrtf(v2[t] + 1e-5f);
    float t2 = (b2[t] - m2[t]) * s2 + bb2[t];
    float* Pl = P + (size_t)layer * 512;
    Pl[c]       = s1;
    Pl[128 + c] = t1;
    Pl[256 + c] = s2;
    Pl[384 + c] = t2;
}

// ---------------------------------------------------------------------------
// h = (1 + eps[layer]) * x   (float4 vectorized; eps read on-device)
// ---------------------------------------------------------------------------
__global__ void gin_init_h(const float* __restrict__ xin,
                           const float* __restrict__ eps, int layer,
                           float* __restrict__ h, long n4) {
    long i = (long)blockIdx.x * blockDim.x + threadIdx.x;
    if (i >= n4) return;
    float oe = 1.0f + eps[layer];
    float4 v = reinterpret_cast<const float4*>(xin)[i];
    v.x *= oe; v.y *= oe; v.z *= oe; v.w *= oe;
    reinterpret_cast<float4*>(h)[i] = v;
}

// ---------------------------------------------------------------------------
// Scatter-add: one wave32 per edge; lane l owns features [4l, 4l+4).
// x fits in L2 (51 MB << 192 MB), so the gather + atomics stay on-chip.
// ---------------------------------------------------------------------------
__global__ void gin_scatter_add(const float* __restrict__ xin,
                                const int* __restrict__ ei,
                                float* __restrict__ h, int E) {
    long gid = (long)blockIdx.x * blockDim.x + threadIdx.x;
    int e = (int)(gid >> 5);
    if (e >= E) return;
    int lane = (int)(gid & 31);
    int src = ei[e];
    int dst = ei[E + e];
    float4 v = reinterpret_cast<const float4*>(xin + (size_t)src * 128)[lane];
    float* hp = h + (size_t)dst * 128 + lane * 4;
    atomicAdd(hp + 0, v.x);
    atomicAdd(hp + 1, v.y);
    atomicAdd(hp + 2, v.z);
    atomicAdd(hp + 3, v.w);
}

// ---------------------------------------------------------------------------
// out[N,128] = act( in[N,128] @ W[128,128] * scale + shift )
// 8 waves / block; each wave computes a 16-row strip with
// v_wmma_f32_16x16x4_f32. W staged in LDS in two 64-column halves by the TDM:
//   tile 128 rows x 64 cols, tensor_dim0_stride=128 (skips other half),
//   pad_enable with 8 DWORDs of pad every 64 DWORDs -> LDS row stride 72.
//
// f32 16x16x4 operand layouts (ISA 7.12.2):
//   A (16x4): lane L -> row M=L%16, K = 2*(L/16) + {vgpr0,vgpr1}
//   B (4x16): lane L -> col N=L%16, K = 2*(L/16) + {vgpr0,vgpr1}
//   C/D     : lane L, vgpr r -> M = r + 8*(L/16), N = L%16
// ---------------------------------------------------------------------------
__global__ __launch_bounds__(256) void gin_gemm128_wmma(
        const float* __restrict__ in, const float* __restrict__ W,
        const float* __restrict__ scale,   // nullable -> 1.0
        const float* __restrict__ shift,
        float* __restrict__ out, int N, int doRelu) {
    __shared__ float lw[128 * LDS_STRIDE];

    const int tid   = threadIdx.x;
    const int lane  = tid & 31;
    const int wave  = tid >> 5;
    const int n16   = lane & 15;   // row (A) / col (B,D) within 16
    const int khalf = lane >> 4;   // which K pair / which M half of D
    const int rowBase = blockIdx.x * 128 + wave * 16;
    const bool active = rowBase < N;

    const int rowA = active ? min(rowBase + n16, N - 1) : 0;
    const float* aRow = in + (size_t)rowA * 128;
    __builtin_prefetch(aRow, 0, 3);            // global_prefetch_b8

    for (int half = 0; half < 2; ++half) {
#if __has_builtin(__builtin_amdgcn_tensor_load_to_lds)
        // ---- TDM staging: W[:, half*64 .. half*64+63] -> padded-row LDS ----
        if (wave == 0) {   // one DMA per block; TDM ignores EXEC, tracked by TENSORcnt
            const unsigned long long ga = (unsigned long long)(size_t)(W + half * 64);
            u32x4 g0;
            g0[0] = 1u;                                        // count=1 (valid user D#)
            g0[1] = gin_lds_off(&lw[0]);                       // lds_addr (bytes)
            g0[2] = (unsigned)(ga & 0xFFFFFFFFu);              // global_addr[31:0]
            g0[3] = (unsigned)((ga >> 32) & 0x1FFFFFFu)        // global_addr[56:32]
                  | (2u << 30);                                // type = 2 ("image")
            i32x8 g1;
            g1[0] = (int)((2u << 16)      // data_size = 4B
                        | (1u << 20)      // pad_enable
                        | (5u << 22)      // pad_interval: every 64 DWORDs
                        | (7u << 25));    // pad_amount: 8 DWORDs  -> row stride 72 floats
            g1[1] = (int)(128u << 16);    // tensor_dim0 = 128 (low16 in [31:16])
            g1[2] = (int)(128u << 16);    // dim0 hi16 = 0 | tensor_dim1 = 128 (low16)
            g1[3] = (int)(64u << 16);     // dim1 hi16 = 0 | tile_dim0 = 64 cols
            g1[4] = 128;                  // tile_dim1 = 128 rows | tile_dim2 = 0
            g1[5] = 128;                  // tensor_dim0_stride = 128 elems (low32)
            g1[6] = (int)(16384u << 16);  // stride0 hi16 = 0 | tensor_dim1_stride low16
            g1[7] = 0;                    // tensor_dim1_stride hi32
            i32x4 gz4;
            gz4[0] = 0; gz4[1] = 0; gz4[2] = 0; gz4[3] = 0;    // 2D tile: groups 2/3 unused
            i32x8 gz8;
            gz8[0] = 0; gz8[1] = 0; gz8[2] = 0; gz8[3] = 0;
            gz8[4] = 0; gz8[5] = 0; gz8[6] = 0; gz8[7] = 0;
            // 6-arg toolchain variant: (g0 x4, g1 x8, x4, x4, x8, cpol)
            __builtin_amdgcn_tensor_load_to_lds(g0, g1, gz4, gz4, gz8, 0);
            __builtin_amdgcn_s_wait_tensorcnt(0);              // s_wait_tensorcnt 0
        }
#else
        // ---- Fallback: manual staging (float4, padded rows) ----
        for (int idx = tid; idx < 128 * 16; idx += 256) {
            int r  = idx >> 4;
            int c4 = (idx & 15) << 2;
            float4 wv = *reinterpret_cast<const float4*>(&W[r * 128 + half * 64 + c4]);
            *reinterpret_cast<float4*>(&lw[r * LDS_STRIDE + c4]) = wv;
        }
#endif
        __syncthreads();

        if (active) {   // wave-uniform guard: EXEC all ones inside WMMA region
            v8f acc[4];
            acc[0] = v8f{}; acc[1] = v8f{}; acc[2] = v8f{}; acc[3] = v8f{};

            for (int k0 = 0; k0 < 128; k0 += 4) {
                const int ka = k0 + khalf * 2;
                v2f a = *reinterpret_cast<const v2f*>(aRow + ka);
                const float* lb0 = &lw[ka * LDS_STRIDE];
                const float* lb1 = &lw[(ka + 1) * LDS_STRIDE];
                v2f b0, b1, b2, b3;
                b0.x = lb0[n16];      b0.y = lb1[n16];
                b1.x = lb0[16 + n16]; b1.y = lb1[16 + n16];
                b2.x = lb0[32 + n16]; b2.y = lb1[32 + n16];
                b3.x = lb0[48 + n16]; b3.y = lb1[48 + n16];
                acc[0] = __builtin_amdgcn_wmma_f32_16x16x4_f32(false, a, false, b0,
                         (short)0, acc[0], false, false);
                acc[1] = __builtin_amdgcn_wmma_f32_16x16x4_f32(false, a, false, b1,
                         (short)0, acc[1], false, false);
                acc[2] = __builtin_amdgcn_wmma_f32_16x16x4_f32(false, a, false, b2,
                         (short)0, acc[2], false, false);
                acc[3] = __builtin_amdgcn_wmma_f32_16x16x4_f32(false, a, false, b3,
                         (short)0, acc[3], false, false);
            }

            // Fused scale/shift/ReLU epilogue.
            #pragma unroll
            for (int t = 0; t < 4; ++t) {
                int col = half * 64 + t * 16 + n16;
                float sc = scale ? scale[col] : 1.0f;
                float sh = shift[col];
                #pragma unroll
                for (int r = 0; r < 8; ++r) {
                    int row = rowBase + r + 8 * khalf;
                    if (row < N) {
                        float v = fmaf(acc[t][r], sc, sh);
                        if (doRelu) v = fmaxf(v, 0.0f);
                        out[(size_t)row * 128 + col] = v;
                    }
                }
            }
        }
        __syncthreads();
    }
}

// ---------------------------------------------------------------------------
// logits = in @ lin2_W[128,10] + b ; out = log_softmax(logits)
// One thread per row; W/b broadcasts hit L1.
// ---------------------------------------------------------------------------
__global__ void gin_lin2_logsoftmax(const float* __restrict__ in,
                                    const float* __restrict__ W,
                                    const float* __restrict__ b,
                                    float* __restrict__ out, int N) {
    int n = blockIdx.x * blockDim.x + threadIdx.x;
    if (n >= N) return;
    float acc[10];
    #pragma unroll
    for (int c = 0; c < 10; ++c) acc[c] = b[c];
    const float* row = in + (size_t)n * 128;
    for (int k = 0; k < 128; ++k) {
        float xv = row[k];
        #pragma unroll
        for (int c = 0; c < 10; ++c) acc[c] = fmaf(xv, W[k * 10 + c], acc[c]);
    }
    float m = acc[0];
    #pragma unroll
    for (int c = 1; c < 10; ++c) m = fmaxf(m, acc[c]);
    float s = 0.0f;
    #pragma unroll
    for (int c = 0; c < 10; ++c) s += expf(acc[c] - m);
    float lse = m + logf(s);
    #pragma unroll
    for (int c = 0; c < 10; ++c) out[(size_t)n * 10 + c] = acc[c] - lse;
}

// ---------------------------------------------------------------------------
extern "C" void kernel_launch(void* const* d_in, const int* in_sizes, int n_in,
                              void* d_out, int out_size, void* d_ws, size_t ws_size,
                              hipStream_t stream) {
    const float* x     = (const float*)d_in[0];
    const int*   ei    = (const int*)d_in[1];     // edge_index, int per harness contract
    const float* eps   = (const float*)d_in[2];
    const float* W1    = (const float*)d_in[3];
    const float* b1    = (const float*)d_in[4];
    const float* g1    = (const float*)d_in[5];
    const float* bb1   = (const float*)d_in[6];
    const float* m1    = (const float*)d_in[7];
    const float* v1    = (const float*)d_in[8];
    const float* W2    = (const float*)d_in[9];
    const float* b2    = (const float*)d_in[10];
    const float* g2    = (const float*)d_in[11];
    const float* bb2   = (const float*)d_in[12];
    const float* m2    = (const float*)d_in[13];
    const float* v2    = (const float*)d_in[14];
    const float* lin1W = (const float*)d_in[15];
    const float* lin1b = (const float*)d_in[16];
    const float* lin2W = (const float*)d_in[17];
    const float* lin2b = (const float*)d_in[18];

    const int N = in_sizes[0] / 128;
    const int E = in_sizes[1] / 2;
    const int L = in_sizes[2];

    float* B0 = (float*)d_ws;
    float* B1 = B0 + (size_t)N * 128;
    float* P  = B1 + (size_t)N * 128;   // L * 512 folded BN params

    const int LH = L * 128;
    gin_fold_bn<<<(LH + 255) / 256, 256, 0, stream>>>(b1, g1, bb1, m1, v1,
                                                      b2, g2, bb2, m2, v2, P, LH);

    const long n4    = (long)N * 32;                  // float4 count
    const int  gInit = (int)((n4 + 255) / 256);
    const int  gScat = (int)(((long)E * 32 + 255) / 256);
    const int  gGemm = (N + 127) / 128;

    const float* xc = x;
    float* hbuf = B0;
    float* tbuf = B1;
    for (int i = 0; i < L; ++i) {
        gin_init_h<<<gInit, 256, 0, stream>>>(xc, eps, i, hbuf, n4);
        gin_scatter_add<<<gScat, 256, 0, stream>>>(xc, ei, hbuf, E);
        const float* Pl = P + (size_t)i * 512;
        gin_gemm128_wmma<<<gGemm, 256, 0, stream>>>(hbuf, W1 + (size_t)i * 128 * 128,
                                                    Pl, Pl + 128, tbuf, N, 1);
        gin_gemm128_wmma<<<gGemm, 256, 0, stream>>>(tbuf, W2 + (size_t)i * 128 * 128,
                                                    Pl + 256, Pl + 384, hbuf, N, 1);
        xc = hbuf;
        float* tmp = hbuf; hbuf = tbuf; tbuf = tmp;
    }

    // lin1 (+ReLU): scale=1 (nullptr), shift=lin1_b; hbuf != xc is free here.
    gin_gemm128_wmma<<<gGemm, 256, 0, stream>>>(xc, lin1W, nullptr, lin1b, hbuf, N, 1);
    gin_lin2_logsoftmax<<<(N + 255) / 256, 256, 0, stream>>>(hbuf, lin2W, lin2b,
                                                             (float*)d_out, N);
}